// FernBitWord_44779329028742
// MI455X (gfx1250) — compile-verified
//
#include <hip/hip_runtime.h>
#include <stdint.h>

// Problem constants (fixed by the reference).
#define N_IMG   16
#define D_CH    3
#define H_IN    160
#define W_IN    160
#define M_F     8
#define K_F     10
#define MK      80
#define PM      7          // (P-1)/2 with P=15
#define OUT_H   146
#define OUT_W   146

// Tiling.
#define TY          16               // output rows per block
#define RROWS       30               // staged input rows = TY + 14 (sy in [0,13] + bilinear +1)
#define LDS_STRIDE  164              // floats per staged row (16B aligned, >160 -> zero pad via TDM OOB)
#define PLANE_ELEMS (RROWS * LDS_STRIDE)
#define FG          4                // feature groups
#define FPG         (MK / FG)        // 20 features per group
#define NTHREADS    256

typedef unsigned int u32;
typedef __attribute__((ext_vector_type(4))) u32 v4u;
typedef __attribute__((ext_vector_type(8))) int v8i;
typedef __attribute__((ext_vector_type(4))) int v4i;

__device__ __forceinline__ u32 lds_byte_offset(const void* p) {
  // generic -> LDS address space -> 32-bit LDS byte offset
  return (u32)(size_t)(__attribute__((address_space(3))) const char*)p;
}

__global__ __launch_bounds__(NTHREADS) void fern_bits_kernel(
    const float* __restrict__ T,   const float* __restrict__ dx1,
    const float* __restrict__ dx2, const float* __restrict__ dy1,
    const float* __restrict__ dy2, const float* __restrict__ th,
    const float* __restrict__ amb, const int*   __restrict__ channels,
    float* __restrict__ out)
{
  __shared__ float lds[D_CH * PLANE_ELEMS];   // 3 x 30 x 164 floats = 59,040 B
  __shared__ float prm[FPG * 16];             // per-feature params, 1,280 B

  const int n   = blockIdx.x;
  const int y0  = blockIdx.y * TY;
  const int grp = blockIdx.z;
  const int tid = threadIdx.x;

  // ---------------- per-feature parameter precompute ----------------
  if (tid < FPG) {
    const int j = grp * FPG + tid;       // feature index in [0, MK)
    const int m = j / K_F;
    const int k = j - m * K_F;
    const float a1 = dx1[j], a2 = dx2[j], b1 = dy1[j], b2 = dy2[j];
    const float fl_a1 = floorf(a1), fl_a2 = floorf(a2);
    const float fl_b1 = floorf(b1), fl_b2 = floorf(b2);
    const float fx1 = fabsf(a1 - fl_a1), fx2 = fabsf(a2 - fl_a2);
    const float fy1 = fabsf(b1 - fl_b1), fy2 = fabsf(b2 - fl_b2);
    const int sx1 = PM + (int)fl_a1, sx2 = PM + (int)fl_a2;
    const int sy1 = PM + (int)fl_b1, sy2 = PM + (int)fl_b2;
    const float pos = amb[(m * 2 + 0) * K_F + k];
    const float neg = amb[(m * 2 + 1) * K_F + k];
    float* p = &prm[tid * 16];
    p[0]  = (float)channels[k];          // ch = channels[j % K]
    p[1]  = (float)sx1;  p[2] = (float)sy1;
    p[3]  = (float)sx2;  p[4] = (float)sy2;
    p[5]  = fx1; p[6] = fy1;
    p[7]  = fx2; p[8] = fy2;
    p[9]  = th[j] + neg;                 // fused threshold: (temp - th - neg)
    p[10] = 1.0f / (pos - neg + 1e-29f); // scale
  }

  // ---------------- stage input rows [y0, y0+RROWS) of all 3 planes into LDS ----
#if __has_builtin(__builtin_amdgcn_tensor_load_to_lds) && __has_builtin(__builtin_amdgcn_s_wait_tensorcnt)
  if (tid < 32) {  // wave 0 issues the 3 TDM descriptors (TDM ignores EXEC; branch is per-wave)
    #pragma unroll
    for (int c = 0; c < D_CH; ++c) {
      const uint64_t gaddr =
          (uint64_t)(uintptr_t)(T + ((size_t)(n * D_CH + c) * H_IN + (size_t)y0) * W_IN);
      const u32 laddr = lds_byte_offset(&lds[c * PLANE_ELEMS]);

      // D# group 0: count=1, lds_addr, global_addr, type=2 ("image")
      const u32 g0w0 = 1u;
      const u32 g0w1 = laddr;
      const u32 g0w2 = (u32)(gaddr & 0xFFFFFFFFu);
      const u32 g0w3 = (u32)((gaddr >> 32) & 0x1FFFFFFu) | (2u << 30);
      const v4u g0 = (v4u){g0w0, g0w1, g0w2, g0w3};

      // D# group 1: data_size=4B; tensor 160 x (160-y0); tile 164 x 30; stride0=160.
      // tile_dim0 > tensor_dim0 -> OOB columns (incl. zero-pad col) load as 0;
      // rows past the image bottom are OOB in dim1 -> 0.   Gives stride-164 rows.
      const u32 tdim0 = (u32)W_IN;
      const u32 tdim1 = (u32)(H_IN - y0);
      const u32 tile0 = (u32)LDS_STRIDE;
      const u32 tile1 = (u32)RROWS;
      const uint64_t s0 = (uint64_t)W_IN;  // dim0 stride (elements)
      const uint64_t s1 = 0;               // dim1 stride unused (2D tile)
      const u32 g1w0 = (2u << 16);                                   // data_size=2 (4B)
      const u32 g1w1 = (tdim0 & 0xFFFFu) << 16;                      // tensor_dim0[15:0]
      const u32 g1w2 = (tdim0 >> 16) | ((tdim1 & 0xFFFFu) << 16);    // dim0 hi | dim1 lo
      const u32 g1w3 = (tdim1 >> 16) | (tile0 << 16);                // dim1 hi | tile_dim0
      const u32 g1w4 = tile1;                                        // tile_dim1 | tile_dim2=0
      const u32 g1w5 = (u32)(s0 & 0xFFFFFFFFu);
      const u32 g1w6 = (u32)((s0 >> 32) & 0xFFFFu) | ((u32)(s1 & 0xFFFFu) << 16);
      const u32 g1w7 = (u32)(s1 >> 16);
      const v8i g1 = (v8i){(int)g1w0, (int)g1w1, (int)g1w2, (int)g1w3,
                           (int)g1w4, (int)g1w5, (int)g1w6, (int)g1w7};
      const v4i z4 = (v4i){0, 0, 0, 0};                      // groups 2/3 unused (2D)
      const v8i z8 = (v8i){0, 0, 0, 0, 0, 0, 0, 0};          // extra group (clang-23 6-arg form)
      __builtin_amdgcn_tensor_load_to_lds(g0, g1, z4, z4, z8, 0);
    }
    __builtin_amdgcn_s_wait_tensorcnt(0);
  }
#else
  for (int idx = tid; idx < D_CH * PLANE_ELEMS; idx += NTHREADS) {
    const int c   = idx / PLANE_ELEMS;
    const int rem = idx - c * PLANE_ELEMS;
    const int wr  = rem / LDS_STRIDE;
    const int col = rem - wr * LDS_STRIDE;
    const int gr  = y0 + wr;
    float v = 0.0f;
    if (gr < H_IN && col < W_IN)
      v = T[((size_t)(n * D_CH + c) * H_IN + gr) * W_IN + col];
    lds[idx] = v;
  }
#endif
  __syncthreads();

  // ---------------- compute ----------------
  const int validRows = (OUT_H - y0) < TY ? (OUT_H - y0) : TY;
  const int npix = validRows * OUT_W;

  for (int jj = 0; jj < FPG; ++jj) {
    const float* p = &prm[jj * 16];
    const int   ch  = (int)p[0];
    const int   sx1 = (int)p[1], sy1 = (int)p[2];
    const int   sx2 = (int)p[3], sy2 = (int)p[4];
    const float fx1 = p[5], fy1 = p[6], fx2 = p[7], fy2 = p[8];
    const float thr = p[9], scl = p[10];
    const float omfx1 = 1.0f - fx1, omfy1 = 1.0f - fy1;
    const float omfx2 = 1.0f - fx2, omfy2 = 1.0f - fy2;
    const float* plane = &lds[ch * PLANE_ELEMS];

    const int j = grp * FPG + jj;
    float* op = out + ((size_t)(n * MK + j) * OUT_H + (size_t)y0) * OUT_W;

    for (int pix = tid; pix < npix; pix += NTHREADS) {
      const int y = pix / OUT_W;
      const int x = pix - y * OUT_W;

      const float* q1 = plane + (sy1 + y) * LDS_STRIDE + (sx1 + x);
      const float* q2 = plane + (sy2 + y) * LDS_STRIDE + (sx2 + x);

      const float g00a = q1[0],           g01a = q1[1];
      const float g10a = q1[LDS_STRIDE],  g11a = q1[LDS_STRIDE + 1];
      const float g00b = q2[0],           g01b = q2[1];
      const float g10b = q2[LDS_STRIDE],  g11b = q2[LDS_STRIDE + 1];

      const float P1 = omfy1 * (omfx1 * g00a + fx1 * g01a)
                     +   fy1 * (omfx1 * g10a + fx1 * g11a);
      const float P2 = omfy2 * (omfx2 * g00b + fx2 * g01b)
                     +   fy2 * (omfx2 * g10b + fx2 * g11b);

      float t = P1 - P2;
      t = (fabsf(t) < 1e-5f) ? 0.0f : t;
      float r = (t - thr) * scl;
      r = fminf(fmaxf(r, 0.0f), 1.0f);
      __builtin_nontemporal_store(r, op + y * OUT_W + x);
    }
  }
}

extern "C" void kernel_launch(void* const* d_in, const int* in_sizes, int n_in,
                              void* d_out, int out_size, void* d_ws, size_t ws_size,
                              hipStream_t stream) {
  (void)in_sizes; (void)n_in; (void)out_size; (void)d_ws; (void)ws_size;
  const float* T        = (const float*)d_in[0];
  const float* dx1      = (const float*)d_in[1];
  const float* dx2      = (const float*)d_in[2];
  const float* dy1      = (const float*)d_in[3];
  const float* dy2      = (const float*)d_in[4];
  const float* th       = (const float*)d_in[5];
  const float* amb      = (const float*)d_in[6];
  const int*   channels = (const int*)d_in[7];
  // d_in[8] = patch_size (compile-time constant 15 here)
  float* out = (float*)d_out;

  dim3 grid(N_IMG, (OUT_H + TY - 1) / TY, FG);   // 16 x 10 x 4 = 640 blocks
  fern_bits_kernel<<<grid, NTHREADS, 0, stream>>>(T, dx1, dx2, dy1, dy2, th,
                                                  amb, channels, out);
}